// MsaHmmCell_35124242546873
// MI455X (gfx1250) — compile-verified
//
#include <hip/hip_runtime.h>
#include <hip/hip_bf16.h>

#define QN     1027      // number of HMM states (2*512+3)
#define QP     1040      // padded to multiple of 16 (zero-padded rows/cols)
#define BATCH  256
#define LSEQ   512
#define LMOD   512
#define SALPH  25
#define HMM_EPS 1e-32f

#define KCHUNK 520       // K staged into LDS in two chunks (520*2 == QP)
#define LDSTR  532       // padded LDS row stride in floats (bank-conflict dodge)

typedef __attribute__((ext_vector_type(2))) float v2f;
typedef __attribute__((ext_vector_type(8))) float v8f;

#if __has_builtin(__builtin_amdgcn_global_load_async_to_lds_b32)
#define HAVE_ASYNC_LDS 1
typedef __attribute__((address_space(1))) int gas_int;   // global (AS1) int
typedef __attribute__((address_space(3))) int las_int;   // LDS (AS3) int
#else
#define HAVE_ASYNC_LDS 0
#endif

// ---------------- zero the per-step scale accumulators -----------------------
__global__ void zero_kernel(float* s_all, int n_s) {
    int i = blockIdx.x * blockDim.x + threadIdx.x;
    if (i < n_s) s_all[i] = 0.f;
}

// ---------------- A = softmax(trans_logits, axis=-1), zero-padded to QPxQP ----
__global__ __launch_bounds__(256) void softmax_rows_kernel(const float* __restrict__ logits,
                                                           float* __restrict__ Apad) {
    __shared__ float red[256];
    const int row = blockIdx.x;
    const int tid = threadIdx.x;
    if (row >= QN) {
        for (int c = tid; c < QP; c += 256) Apad[(size_t)row * QP + c] = 0.f;
        return;
    }
    const float* lrow = logits + (size_t)row * QN;
    float mx = -3.4e38f;
    for (int c = tid; c < QN; c += 256) mx = fmaxf(mx, lrow[c]);
    red[tid] = mx; __syncthreads();
    for (int off = 128; off > 0; off >>= 1) {
        if (tid < off) red[tid] = fmaxf(red[tid], red[tid + off]);
        __syncthreads();
    }
    const float m = red[0];
    __syncthreads();
    float sum = 0.f;
    for (int c = tid; c < QN; c += 256) sum += expf(lrow[c] - m);
    red[tid] = sum; __syncthreads();
    for (int off = 128; off > 0; off >>= 1) {
        if (tid < off) red[tid] += red[tid + off];
        __syncthreads();
    }
    const float inv = 1.f / red[0];
    for (int c = tid; c < QP; c += 256)
        Apad[(size_t)row * QP + c] = (c < QN) ? expf(lrow[c] - m) * inv : 0.f;
}

// ---------------- init = softmax(init_logits), zero-padded to QP -------------
__global__ __launch_bounds__(256) void softmax_init_kernel(const float* __restrict__ logits,
                                                           float* __restrict__ initpad) {
    __shared__ float red[256];
    const int tid = threadIdx.x;
    float mx = -3.4e38f;
    for (int c = tid; c < QN; c += 256) mx = fmaxf(mx, logits[c]);
    red[tid] = mx; __syncthreads();
    for (int off = 128; off > 0; off >>= 1) {
        if (tid < off) red[tid] = fmaxf(red[tid], red[tid + off]);
        __syncthreads();
    }
    const float m = red[0];
    __syncthreads();
    float sum = 0.f;
    for (int c = tid; c < QN; c += 256) sum += expf(logits[c] - m);
    red[tid] = sum; __syncthreads();
    for (int off = 128; off > 0; off >>= 1) {
        if (tid < off) red[tid] += red[tid + off];
        __syncthreads();
    }
    const float inv = 1.f / red[0];
    for (int c = tid; c < QP; c += 256)
        initpad[c] = (c < QN) ? expf(logits[c] - m) * inv : 0.f;
}

// --------- Bt[c][q] = softmax emission prob of symbol c in state q -----------
__global__ void emission_kernel(const float* __restrict__ em,
                                const float* __restrict__ ins,
                                float* __restrict__ Bt) {
    const int q = blockIdx.x * blockDim.x + threadIdx.x;
    if (q >= QP) return;
    if (q >= QN - 1) {                       // terminal state (emits only sym 25) or padding
        for (int c = 0; c < SALPH; ++c) Bt[(size_t)c * QP + q] = 0.f;
        return;
    }
    const float* rowp = (q == 0) ? ins
                       : ((q <= LMOD) ? em + (size_t)(q - 1) * SALPH : ins);
    float v[SALPH];
    float mx = -3.4e38f;
    for (int c = 0; c < SALPH; ++c) { v[c] = rowp[c]; mx = fmaxf(mx, v[c]); }
    float s = 0.f;
    for (int c = 0; c < SALPH; ++c) { v[c] = expf(v[c] - mx); s += v[c]; }
    const float inv = 1.f / s;
    for (int c = 0; c < SALPH; ++c) Bt[(size_t)c * QP + q] = v[c] * inv;
}

// --------- t=0: alpha0[b][q] = init[q] * Bt[seq[b][0]][q] (raw), s0[b] = sum --
__global__ __launch_bounds__(256) void init_alpha_kernel(const float* __restrict__ initpad,
                                                         const float* __restrict__ Bt,
                                                         const int*   __restrict__ seq,
                                                         float* __restrict__ alpha,
                                                         float* __restrict__ s0) {
    __shared__ float red[256];
    const int b = blockIdx.x, tid = threadIdx.x;
    const int sym = seq[(size_t)b * LSEQ];
    const float* bt = Bt + (size_t)sym * QP;
    float partial = 0.f;
    for (int q = tid; q < QP; q += 256) {
        float v = initpad[q] * bt[q];
        alpha[(size_t)b * QP + q] = v;
        partial += v;
    }
    red[tid] = partial; __syncthreads();
    for (int off = 128; off > 0; off >>= 1) {
        if (tid < off) red[tid] += red[tid + off];
        __syncthreads();
    }
    if (tid == 0) s0[b] = red[0];
}

// --------- one forward step: out = ((alpha_raw/s_prev) @ A) * E_t ------------
// 16 waves per WG; WG owns one 16-col n-panel whose A columns are staged
// (transposed, K-chunked) into LDS with GLOBAL_LOAD_ASYNC_TO_LDS_B32
// (direct memory->LDS, ASYNCcnt-tracked) and shared by all 16 m-tiles.
// Inner loop: 1 global b64 (alpha) + 1/2 ds_load_2addr_b64 (panel) + 1 f32 WMMA.
// 1/s_prev is folded into the epilogue (linearity of the transition map).
__global__ __launch_bounds__(512) void hmm_step_wmma_kernel(
        const float* __restrict__ alpha_in,   // BATCH x QP (raw from prev step)
        float*       __restrict__ alpha_out,  // BATCH x QP (raw)
        const float* __restrict__ Apad,       // QP x QP
        const float* __restrict__ Bt,         // SALPH x QP
        const int*   __restrict__ seq,        // BATCH x LSEQ
        const float* __restrict__ s_prev,     // BATCH scales from prev step
        float*       __restrict__ s_t,        // BATCH accumulators (this step)
        int t) {
    __shared__ float Bp[16 * LDSTR];          // Bp[n][k] = A[kc+k][n0+n]

    const int tid  = threadIdx.x;             // 0..511
    const int n0   = blockIdx.x * 16;         // state-col panel
    const int wave = tid >> 5;                // 0..15 -> m-tile
    const int lane = tid & 31;
    const int half = lane >> 4;               // 0: lanes 0-15, 1: lanes 16-31
    const int l16  = lane & 15;
    const int m0   = wave * 16;               // batch-row tile

    v8f acc = {};
    const float* aptr = alpha_in + (size_t)(m0 + l16) * QP + (half << 1);
    const float* bptr = Bp + (size_t)l16 * LDSTR + (half << 1);

    for (int kc = 0; kc < QP; kc += KCHUNK) {
        __syncthreads();                      // protect LDS reuse across chunks
        // stage A[kc:kc+KCHUNK, n0:n0+16] transposed into LDS
        for (int idx = tid; idx < 16 * KCHUNK; idx += 512) {
            int k = idx >> 4;                 // 0..KCHUNK-1
            int n = idx & 15;
#if HAVE_ASYNC_LDS
            __builtin_amdgcn_global_load_async_to_lds_b32(
                (gas_int*)(Apad + (size_t)(kc + k) * QP + n0 + n),
                (las_int*)&Bp[n * LDSTR + k], 0, 0);
#else
            Bp[n * LDSTR + k] = Apad[(size_t)(kc + k) * QP + n0 + n];
#endif
        }
#if HAVE_ASYNC_LDS
        asm volatile("s_wait_asynccnt 0x0" ::: "memory");  // own transfers done
#endif
        __syncthreads();                      // everyone's transfers visible

        for (int k = 0; k < KCHUNK; k += 4) {
            v2f a = *(const v2f*)(aptr + kc + k);   // 8B-aligned global b64
            v2f b = *(const v2f*)(bptr + k);        // 8B-aligned ds b64
            acc = __builtin_amdgcn_wmma_f32_16x16x4_f32(
                      false, a, false, b, (short)0, acc, false, false);
        }
    }

    const int n = n0 + l16;
#pragma unroll
    for (int i = 0; i < 8; ++i) {
        const int row = m0 + i + (half << 3);
        const float inv = 1.f / fmaxf(s_prev[row], HMM_EPS);
        const int sym = seq[(size_t)row * LSEQ + t];
        float v = acc[i] * inv * Bt[(size_t)sym * QP + n]; // E_t (0 for pads/terminal)
        alpha_out[(size_t)row * QP + n] = v;
        float sum = v;                                      // 16-lane tile row sum
#pragma unroll
        for (int off = 1; off < 16; off <<= 1) sum += __shfl_xor(sum, off, 16);
        if (l16 == 0) atomicAdd(&s_t[row], sum);
    }
}

// --------- ll[b] = sum_t log(max(s_t[b], eps)) -------------------------------
__global__ void ll_kernel(const float* __restrict__ s_all, float* __restrict__ ll) {
    int b = blockIdx.x * blockDim.x + threadIdx.x;
    if (b >= BATCH) return;
    float acc = 0.f;
    for (int t = 0; t < LSEQ; ++t)
        acc += logf(fmaxf(s_all[(size_t)t * BATCH + b], HMM_EPS));
    ll[b] = acc;
}

extern "C" void kernel_launch(void* const* d_in, const int* in_sizes, int n_in,
                              void* d_out, int out_size, void* d_ws, size_t ws_size,
                              hipStream_t stream) {
    const int*   seq   = (const int*)  d_in[0];   // (256, 512) int32
    const float* em    = (const float*)d_in[1];   // (512, 25)
    const float* ins   = (const float*)d_in[2];   // (25,)
    const float* trans = (const float*)d_in[3];   // (1027, 1027)
    const float* initl = (const float*)d_in[4];   // (1027,)
    float* ll = (float*)d_out;                    // (256,)

    char* ws = (char*)d_ws;
    size_t off = 0;
    float* Apad    = (float*)(ws + off); off += (size_t)QP * QP * 4;
    float* Bt      = (float*)(ws + off); off += (size_t)SALPH * QP * 4;
    float* initpad = (float*)(ws + off); off += (size_t)QP * 4;
    off = (off + 255) & ~(size_t)255;
    float* alpha0  = (float*)(ws + off); off += (size_t)BATCH * QP * 4;
    float* alpha1  = (float*)(ws + off); off += (size_t)BATCH * QP * 4;
    float* s_all   = (float*)(ws + off); off += (size_t)LSEQ * BATCH * 4;

    const int n_s = LSEQ * BATCH;
    zero_kernel<<<(n_s + 255) / 256, 256, 0, stream>>>(s_all, n_s);
    softmax_rows_kernel<<<QP, 256, 0, stream>>>(trans, Apad);
    emission_kernel<<<(QP + 255) / 256, 256, 0, stream>>>(em, ins, Bt);
    softmax_init_kernel<<<1, 256, 0, stream>>>(initl, initpad);

    init_alpha_kernel<<<BATCH, 256, 0, stream>>>(initpad, Bt, seq, alpha0, s_all);

    float* bufs[2] = { alpha0, alpha1 };
    for (int t = 1; t < LSEQ; ++t) {
        float* prev = bufs[(t + 1) & 1];
        float* cur  = bufs[t & 1];
        hmm_step_wmma_kernel<<<dim3(QP / 16), 512, 0, stream>>>(
            prev, cur, Apad, Bt, seq,
            s_all + (size_t)(t - 1) * BATCH,      // scales of previous step
            s_all + (size_t)t * BATCH, t);
        // no per-step normalize kernel: 1/s folded into next step's epilogue
    }

    ll_kernel<<<1, BATCH, 0, stream>>>(s_all, ll);
}